// RNN_87351044866483
// MI455X (gfx1250) — compile-verified
//
#include <hip/hip_runtime.h>
#include <hip/hip_bf16.h>
#include <math.h>

// ---------------------------------------------------------------------------
// RNN (degenerate: timesteps independent, only last timestep reaches output).
//   x   = emb[X[:, -1]]                 [64, 1024]
//   h   = tanh(x @ W_xh + b_h)          [64, 2048]
//   4x: h = tanh(h @ W_hh + b_h)        [64, 2048]
//   out = h @ W_hy + b_y                [64, 32000]
// HBM-bound (~290 MB weight reads @ 23.3 TB/s ~ 12.5 us). fp32 WMMA
// (v_wmma_f32_16x16x4_f32). K/N are template constants so all row strides
// fold into 24-bit instruction offsets (no per-iteration 64-bit address math).
// ---------------------------------------------------------------------------

typedef __attribute__((ext_vector_type(2))) float v2f;
typedef __attribute__((ext_vector_type(8))) float v8f;

#define BATCH  64
#define SEQ    512
#define EMBED  1024
#define HIDDEN 2048
#define VOCAB  32000

// -------- gather last-timestep embeddings: xlast[b, :] = emb[X[b, SEQ-1], :]
__global__ __launch_bounds__(256)
void gather_last_kernel(const int* __restrict__ X,
                        const float* __restrict__ emb,
                        float* __restrict__ xlast) {
  const int b    = blockIdx.x;
  const long row = (long)X[b * SEQ + (SEQ - 1)];
  const float4* src = (const float4*)(emb + row * EMBED);
  float4*       dst = (float4*)(xlast + (long)b * EMBED);
  dst[threadIdx.x] = src[threadIdx.x];   // 256 threads * 16B = 1024 floats
}

// -------- C[64,N] = act(A[64,K] @ W[K,N] + bias), act = tanh or identity.
// Block = 128 threads = 4 waves. Wave w computes the 16x16 tile at
// rows [16w, 16w+16), cols [16*blockIdx.x, ...+16). Full M=64 per block so
// each W column-strip streams from HBM exactly once (4-wave reuse in WGP$).
//
// v_wmma_f32_16x16x4_f32 lane layouts (ISA 7.12.2, 32-bit A 16x4 / C 16x16):
//   A: lane l<16 holds (M=l, K={0,1}); lane 16+l holds (M=l, K={2,3})
//   B: lane l<16 holds (N=l, K={0,1}); lane 16+l holds (N=l, K={2,3})
//   C: VGPR r, lanes 0-15 -> (M=r,   N=lane); lanes 16-31 -> (M=r+8, N=lane-16)
template <bool TANH, int K, int N>
__global__ __launch_bounds__(128)
void gemm_wmma_f32(const float* __restrict__ A,     // [64 x K] row-major
                   const float* __restrict__ W,     // [K  x N] row-major
                   const float* __restrict__ bias,  // [N]
                   float* __restrict__ C) {         // [64 x N]
  const int wave = threadIdx.x >> 5;   // 0..3 -> m-tile
  const int lane = threadIdx.x & 31;
  const int half = lane >> 4;          // 0: K pair {0,1}, 1: K pair {2,3}
  const int l    = lane & 15;
  const int n0   = blockIdx.x * 16;
  const int m0   = wave * 16;

  const float* Arow = A + (m0 + l) * K + 2 * half;     // +k      -> K pair
  const float* Wcol = W + (2 * half) * N + (n0 + l);   // +k*N    -> K pair

  v8f acc0 = {}, acc1 = {}, acc2 = {}, acc3 = {};

#pragma unroll 2
  for (int k = 0; k < K; k += 16) {
    // K rows k..k+3
    v2f a0 = *(const v2f*)(Arow + k);
    v2f b0;
    b0.x = Wcol[(k + 0) * N];
    b0.y = Wcol[(k + 1) * N];
    acc0 = __builtin_amdgcn_wmma_f32_16x16x4_f32(
        false, a0, false, b0, (short)0, acc0, false, false);

    // K rows k+4..k+7
    v2f a1 = *(const v2f*)(Arow + k + 4);
    v2f b1;
    b1.x = Wcol[(k + 4) * N];
    b1.y = Wcol[(k + 5) * N];
    acc1 = __builtin_amdgcn_wmma_f32_16x16x4_f32(
        false, a1, false, b1, (short)0, acc1, false, false);

    // K rows k+8..k+11
    v2f a2 = *(const v2f*)(Arow + k + 8);
    v2f b2;
    b2.x = Wcol[(k + 8) * N];
    b2.y = Wcol[(k + 9) * N];
    acc2 = __builtin_amdgcn_wmma_f32_16x16x4_f32(
        false, a2, false, b2, (short)0, acc2, false, false);

    // K rows k+12..k+15
    v2f a3 = *(const v2f*)(Arow + k + 12);
    v2f b3;
    b3.x = Wcol[(k + 12) * N];
    b3.y = Wcol[(k + 13) * N];
    acc3 = __builtin_amdgcn_wmma_f32_16x16x4_f32(
        false, a3, false, b3, (short)0, acc3, false, false);
  }

  v8f acc = (acc0 + acc1) + (acc2 + acc3);

  const float bval = bias[n0 + l];
  float* Cout = C + n0 + l;
#pragma unroll
  for (int r = 0; r < 8; ++r) {
    const int row = m0 + r + 8 * half;
    float v = acc[r] + bval;
    if (TANH) v = tanhf(v);
    Cout[row * N] = v;   // row*N*4 <= 8.06 MB: fits signed-24 inst offset
  }
}

extern "C" void kernel_launch(void* const* d_in, const int* in_sizes, int n_in,
                              void* d_out, int out_size, void* d_ws, size_t ws_size,
                              hipStream_t stream) {
  const int*   X    = (const int*)d_in[0];    // [64, 512]
  const float* emb  = (const float*)d_in[1];  // [32000, 1024]
  const float* W_xh = (const float*)d_in[2];  // [1024, 2048]
  const float* W_hh = (const float*)d_in[3];  // [2048, 2048]
  const float* W_hy = (const float*)d_in[4];  // [2048, 32000]
  const float* b_h  = (const float*)d_in[5];  // [2048]
  const float* b_y  = (const float*)d_in[6];  // [32000]
  float* out = (float*)d_out;                 // [64, 32000]

  // Workspace layout (1.31 MB total)
  float* xlast = (float*)d_ws;                 // 64*1024 = 256 KB
  float* h_a   = xlast + BATCH * EMBED;        // 64*2048 = 512 KB
  float* h_b   = h_a + BATCH * HIDDEN;         // 64*2048 = 512 KB

  // 1) gather last-timestep embeddings
  gather_last_kernel<<<BATCH, 256, 0, stream>>>(X, emb, xlast);

  // 2) input layer: h = tanh(x @ W_xh + b_h)
  gemm_wmma_f32<true, EMBED, HIDDEN><<<HIDDEN / 16, 128, 0, stream>>>(
      xlast, W_xh, b_h, h_a);

  // 3) 4 hidden layers: h = tanh(h @ W_hh + b_h), ping-pong h_a <-> h_b
  gemm_wmma_f32<true, HIDDEN, HIDDEN><<<HIDDEN / 16, 128, 0, stream>>>(h_a, W_hh, b_h, h_b);
  gemm_wmma_f32<true, HIDDEN, HIDDEN><<<HIDDEN / 16, 128, 0, stream>>>(h_b, W_hh, b_h, h_a);
  gemm_wmma_f32<true, HIDDEN, HIDDEN><<<HIDDEN / 16, 128, 0, stream>>>(h_a, W_hh, b_h, h_b);
  gemm_wmma_f32<true, HIDDEN, HIDDEN><<<HIDDEN / 16, 128, 0, stream>>>(h_b, W_hh, b_h, h_a);

  // 4) output layer (HBM-bound over 262 MB W_hy): out = h @ W_hy + b_y
  gemm_wmma_f32<false, HIDDEN, VOCAB><<<VOCAB / 16, 128, 0, stream>>>(
      h_a, W_hy, b_y, out);
}